// ConvFace_43748536877375
// MI455X (gfx1250) — compile-verified
//
#include <hip/hip_runtime.h>

typedef float v2f __attribute__((ext_vector_type(2)));
typedef float v8f __attribute__((ext_vector_type(8)));

#define C_CH   64
#define KNB    16
#define TILE_M 16
#define WAVES  8
#define BLK    (WAVES * 32)

// ---------------- kernel 0: zero global stat accumulators ----------------
__global__ __launch_bounds__(128) void zero_stats_k(float* __restrict__ stats) {
    stats[threadIdx.x] = 0.0f; // 128 floats: [0..63]=sum, [64..127]=sumsq
}

// ---------------- kernel 1: fea [B][C][F] -> feaT [B][F][C] ----------------
__global__ __launch_bounds__(256) void transpose_k(const float* __restrict__ fea,
                                                   float* __restrict__ feaT, int F) {
    __shared__ float t[64][65]; // padded: conflict-free
    const int b  = blockIdx.y;
    const int f0 = blockIdx.x * 64;
    const float* src = fea  + (size_t)b * C_CH * F;
    float*       dst = feaT + (size_t)b * F * C_CH;
    const int tx = threadIdx.x & 63;
    const int ty = threadIdx.x >> 6; // 0..3
#pragma unroll
    for (int p = 0; p < 16; ++p) {
        int c = ty + p * 4;
        int f = f0 + tx;
        t[c][tx] = (f < F) ? src[(size_t)c * F + f] : 0.0f; // coalesced read
    }
    __syncthreads();
#pragma unroll
    for (int p = 0; p < 16; ++p) {
        int fr = ty + p * 4;
        int f  = f0 + fr;
        if (f < F) dst[(size_t)f * C_CH + tx] = t[tx][fr];   // coalesced write
    }
}

// ---------------- kernel 2: gather-sum + 1x1 conv (WMMA f32) + stats ----------------
__global__ __launch_bounds__(BLK) void fused_gather_gemm_k(
    const float* __restrict__ feaT,   // [B][F][64]
    const int*   __restrict__ ring,   // [B][F][16]
    const float* __restrict__ W,      // [64][64]  (out, in)
    const float* __restrict__ bias,   // [64]
    float*       __restrict__ y,      // [B][64][F] (pre-norm, to d_out)
    float*       __restrict__ gstats, // [128]
    int F, int tilesPerBatch, int numTiles)
{
    __shared__ float sW[C_CH * C_CH];                  // 16 KB
    __shared__ float sSum[WAVES][TILE_M * C_CH];       // 8 x 4 KB  gathered tiles
    __shared__ float sTile[WAVES][TILE_M * 17];        // padded D-tile transpose
    __shared__ float sStats[2 * C_CH];

    const int tid  = threadIdx.x;
    const int wave = tid >> 5;
    const int lane = tid & 31;
    const int l    = lane & 15;     // row/col within 16
    const int hi   = lane >> 4;     // 0 or 1 (K-pair select for WMMA A/B)

    for (int i = tid; i < C_CH * C_CH; i += BLK) sW[i] = W[i];
    if (tid < 2 * C_CH) sStats[tid] = 0.0f;
    __syncthreads();

    const int tileRaw = blockIdx.x * WAVES + wave;
    const float statw = (tileRaw < numTiles) ? 1.0f : 0.0f; // dup tiles: no stat contribution
    const int tile = (tileRaw < numTiles) ? tileRaw : (numTiles - 1);
    const int b    = tile / tilesPerBatch;
    const int f0   = (tile - b * tilesPerBatch) * TILE_M;

    const float* feaTb = feaT + (size_t)b * F * C_CH;
    const int*   ringb = ring + (size_t)b * F * KNB;

    // ---- gather + neighbor-sum: 16 faces, each lane owns channels {2*lane, 2*lane+1} ----
    float* sumw = sSum[wave];
    for (int i = 0; i < TILE_M; ++i) {
        const int f = f0 + i;
        const int myIdx = ringb[(size_t)f * KNB + l]; // lanes 0..15 hold the 16 indices
        float ax = 0.0f, ay = 0.0f;
#pragma unroll
        for (int k = 0; k < KNB; ++k) {
            const int r = __shfl(myIdx, k, 32);
            const float* p = feaTb + (size_t)r * C_CH + 2 * lane; // 256B/row contiguous
            ax += p[0];
            ay += p[1];
        }
        sumw[i * C_CH + 2 * lane]     = ax;
        sumw[i * C_CH + 2 * lane + 1] = ay;
    }
    __syncthreads();

    // ---- A fragments (32-bit 16x4 A layout): lane<16 -> K=4k+{0,1}, lane>=16 -> K=4k+{2,3} ----
    v2f a[16];
#pragma unroll
    for (int k = 0; k < 16; ++k) {
        const int col = 4 * k + 2 * hi;
        a[k][0] = sumw[l * C_CH + col];
        a[k][1] = sumw[l * C_CH + col + 1];
    }

    float* tilep = sTile[wave];
#pragma unroll 1
    for (int ot = 0; ot < 4; ++ot) {
        v8f acc = {0.f, 0.f, 0.f, 0.f, 0.f, 0.f, 0.f, 0.f};
        const int orow = ot * 16 + l; // output channel for this lane's B column
#pragma unroll
        for (int k = 0; k < 16; ++k) {
            const int col = 4 * k + 2 * hi;
            v2f bm;
            bm[0] = sW[orow * C_CH + col];     // B[k][n] = W[o][c], same pair addressing
            bm[1] = sW[orow * C_CH + col + 1];
            acc = __builtin_amdgcn_wmma_f32_16x16x4_f32(
                false, a[k], false, bm, (short)0, acc, false, false);
        }

        // bias + per-channel stats (lane's column n = l -> channel orow, 8 faces each half)
        const float bo = bias[orow];
        float s = 0.0f, s2 = 0.0f;
#pragma unroll
        for (int j = 0; j < 8; ++j) {
            const float v = acc[j] + bo;
            acc[j] = v;
            s += v;
            s2 += v * v;
        }
        atomicAdd(&sStats[orow], s * statw);
        atomicAdd(&sStats[C_CH + orow], s2 * statw);

        // D tile -> LDS (element m = hi*8+j, n = l), then coalesced float4 stores
#pragma unroll
        for (int j = 0; j < 8; ++j)
            tilep[(hi * 8 + j) * 17 + l] = acc[j];
        __syncthreads();
#pragma unroll
        for (int p = 0; p < 2; ++p) {
            const int ol = (lane >> 2) + p * 8;  // output channel row (0..15)
            const int q  = (lane & 3) * 4;       // face chunk
            float4 v;
            v.x = tilep[(q + 0) * 17 + ol];
            v.y = tilep[(q + 1) * 17 + ol];
            v.z = tilep[(q + 2) * 17 + ol];
            v.w = tilep[(q + 3) * 17 + ol];
            const size_t off = ((size_t)(b * C_CH + ot * 16 + ol)) * F + f0 + q;
            *(float4*)(y + off) = v;
        }
        __syncthreads();
    }

    __syncthreads();
    if (tid < 2 * C_CH) atomicAdd(&gstats[tid], sStats[tid]);
}

// ---------------- kernel 3: finalize BN scale/shift ----------------
__global__ __launch_bounds__(64) void finalize_k(const float* __restrict__ gstats,
                                                 const float* __restrict__ gamma,
                                                 const float* __restrict__ beta,
                                                 float* __restrict__ ss, float invN) {
    const int o = threadIdx.x;
    const float mean = gstats[o] * invN;
    const float var  = gstats[64 + o] * invN - mean * mean;
    const float sc   = gamma[o] * rsqrtf(var + 1e-5f);
    ss[o]      = sc;
    ss[64 + o] = beta[o] - mean * sc;
}

// ---------------- kernel 4: normalize + ReLU (in place on y) ----------------
__global__ __launch_bounds__(256) void norm_relu_k(float* __restrict__ y,
                                                   const float* __restrict__ ss,
                                                   int F4, int n4) {
    const int i = blockIdx.x * 256 + threadIdx.x;
    if (i >= n4) return;
    const int c = (i / F4) & 63;
    const float sc = ss[c], sh = ss[64 + c];
    float4 v = ((float4*)y)[i];
    v.x = fmaxf(v.x * sc + sh, 0.0f);
    v.y = fmaxf(v.y * sc + sh, 0.0f);
    v.z = fmaxf(v.z * sc + sh, 0.0f);
    v.w = fmaxf(v.w * sc + sh, 0.0f);
    ((float4*)y)[i] = v;
}

extern "C" void kernel_launch(void* const* d_in, const int* in_sizes, int n_in,
                              void* d_out, int out_size, void* d_ws, size_t ws_size,
                              hipStream_t stream) {
    const float* fea   = (const float*)d_in[0];
    const int*   ring  = (const int*)d_in[1];   // ring_n indices
    const float* W     = (const float*)d_in[2];
    const float* bias  = (const float*)d_in[3];
    const float* gamma = (const float*)d_in[4];
    const float* beta  = (const float*)d_in[5];
    float* out = (float*)d_out;

    const int B = 4;
    const int F = in_sizes[0] / (B * C_CH); // fea = [B][64][F]

    float* feaT   = (float*)d_ws;                 // B*F*64 floats (~51 MB)
    float* gstats = feaT + (size_t)B * F * C_CH;  // 128 floats
    float* ss     = gstats + 128;                 // 128 floats

    zero_stats_k<<<1, 128, 0, stream>>>(gstats);

    dim3 tg((F + 63) / 64, B);
    transpose_k<<<tg, 256, 0, stream>>>(fea, feaT, F);

    const int tilesPerBatch = (F + TILE_M - 1) / TILE_M;
    const int numTiles = B * tilesPerBatch;
    const int blocks = (numTiles + WAVES - 1) / WAVES;
    fused_gather_gemm_k<<<blocks, BLK, 0, stream>>>(feaT, ring, W, bias, out, gstats,
                                                    F, tilesPerBatch, numTiles);

    finalize_k<<<1, 64, 0, stream>>>(gstats, gamma, beta, ss, 1.0f / ((float)B * (float)F));

    const int n4 = (B * C_CH * F) / 4;
    norm_relu_k<<<(n4 + 255) / 256, 256, 0, stream>>>(out, ss, F / 4, n4);
}